// Conver_loss_90623809946026
// MI455X (gfx1250) — compile-verified
//
#include <hip/hip_runtime.h>
#include <hip/hip_bf16.h>

typedef __attribute__((ext_vector_type(2))) float v2f;
typedef __attribute__((ext_vector_type(8))) float v8f;

#define Hdim 384
#define Wdim 384
#define HWSZ (Hdim * Wdim)
#define Bdim 64
#define INTER 382           // interior points per dimension
#define ROWS_PER_BLOCK 24
#define GRID_Y 16           // ceil(382 / 24)
#define BLOCK_T 384         // 12 waves of 32
#define NPTS ((double)Bdim * (double)INTER * (double)INTER)

__global__ __launch_bounds__(BLOCK_T)
void pde_residual_partial(const float* __restrict__ field,
                          const float* __restrict__ grid,
                          const float* __restrict__ design,
                          float* __restrict__ partials)
{
    const int b   = blockIdx.x;       // batch
    const int by  = blockIdx.y;       // row strip
    const int tid = threadIdx.x;
    const int j   = tid + 1;          // interior column
    const bool active = (tid < INTER);

    const float* __restrict__ P = field + (size_t)(b * 4 + 0) * HWSZ;
    const float* __restrict__ T = field + (size_t)(b * 4 + 1) * HWSZ;
    const float* __restrict__ U = field + (size_t)(b * 4 + 2) * HWSZ;
    const float* __restrict__ V = field + (size_t)(b * 4 + 3) * HWSZ;
    const float* __restrict__ X = grid  + (size_t)(b * 2 + 0) * HWSZ;
    const float* __restrict__ Y = grid  + (size_t)(b * 2 + 1) * HWSZ;

    // ---- nanofluid parameters (per batch, f32 to match reference) ----
    const float per   = design[b * 4 + 3];
    const float rho   = per * 3880.0f + (1.0f - per) * 998.2f;
    const float Cp    = ((1.0f - per) * 998.2f * 4182.0f + per * 3880.0f * 773.0f) / rho;
    const float miu   = 0.000993f * (123.0f * per * per + 7.3f * per + 1.0f);
    const float t1    = (3.0f * per - 1.0f) * 36.0f + (2.0f - 3.0f * per) * 0.597f;
    const float DELTA = t1 * t1 + 8.0f * 36.0f * 0.597f;
    const float lamda = 0.25f * (t1 + sqrtf(DELTA));
    const float inv_rho = 1.0f / rho;
    const float coefT   = lamda / (rho * Cp);

    float sc = 0.0f, su = 0.0f, sv = 0.0f, st = 0.0f;

    const int i0 = 1 + by * ROWS_PER_BLOCK;
    const int i1 = (i0 + ROWS_PER_BLOCK < Hdim - 1) ? (i0 + ROWS_PER_BLOCK) : (Hdim - 1);

    if (active) {
        // 3-row register rotation: m = row i-1, c = row i, p = row i+1
        float pm = P[(i0 - 1) * Wdim + j], pc = P[i0 * Wdim + j];
        float tm = T[(i0 - 1) * Wdim + j], tc = T[i0 * Wdim + j];
        float um = U[(i0 - 1) * Wdim + j], uc = U[i0 * Wdim + j];
        float vm = V[(i0 - 1) * Wdim + j], vc = V[i0 * Wdim + j];

        for (int i = i0; i < i1; ++i) {
            const int rN = (i + 1) * Wdim + j;  // row i+1
            const int rC = i * Wdim + j;        // row i
            const int rP = (i - 1) * Wdim + j;  // row i-1

            // cold loads: next row of the four fields (coalesced along j)
            const float pp = P[rN], tp = T[rN], up = U[rN], vp = V[rN];
            // prefetch two rows ahead -> global_prefetch_b8
            if (i + 2 < Hdim) {
                __builtin_prefetch(&P[rN + Wdim], 0, 3);
                __builtin_prefetch(&T[rN + Wdim], 0, 3);
                __builtin_prefetch(&U[rN + Wdim], 0, 3);
                __builtin_prefetch(&V[rN + Wdim], 0, 3);
            }
            // lateral neighbors (cache hits)
            const float pl = P[rC - 1], pr = P[rC + 1];
            const float tl = T[rC - 1], tr = T[rC + 1];
            const float ul = U[rC - 1], ur = U[rC + 1];
            const float vl = V[rC - 1], vr = V[rC + 1];
            // grid terms: NOTE the reference's dx mixes X(i+1,j) with Y(i-1,j)!
            const float xp1 = X[rN];
            const float ym1 = Y[rP];
            const float yl  = Y[rC - 1], yr = Y[rC + 1];

            const float dxv  = 0.5f * (xp1 - ym1);
            const float dyv  = 0.5f * (yr - yl);
            const float idx  = 1.0f / dxv;
            const float idy  = 1.0f / dyv;
            const float idx2 = idx * idx;
            const float idy2 = idy * idy;

            const float dU_dx = (up - um) * idx, dU_dy = (ur - ul) * idy;
            const float dV_dx = (vp - vm) * idx, dV_dy = (vr - vl) * idy;
            const float dP_dx = (pp - pm) * idx, dP_dy = (pr - pl) * idy;
            const float dT_dx = (tp - tm) * idx, dT_dy = (tr - tl) * idy;

            const float ddUx = (up - 2.0f * uc + um) * idx2, ddUy = (ur - 2.0f * uc + ul) * idy2;
            const float ddVx = (vp - 2.0f * vc + vm) * idx2, ddVy = (vr - 2.0f * vc + vl) * idy2;
            const float ddTx = (tp - 2.0f * tc + tm) * idx2, ddTy = (tr - 2.0f * tc + tl) * idy2;

            sc += dU_dx + dV_dy;
            su += uc * dU_dx + vc * dU_dy + dP_dx * inv_rho - miu * (ddUx + ddUy);
            sv += uc * dV_dx + vc * dV_dy + dP_dy * inv_rho - miu * (ddVx + ddVy);
            st += uc * dT_dx + vc * dT_dy + coefT * (ddTx + ddTy);

            pm = pc; pc = pp;  tm = tc; tc = tp;
            um = uc; uc = up;  vm = vc; vc = vp;
        }
    }

    // ---- block reduction: stage partials, then WMMA-sum with all-ones B ----
    __shared__ float part[4][BLOCK_T];
    part[0][tid] = sc; part[1][tid] = su; part[2][tid] = sv; part[3][tid] = st;
    __syncthreads();

    const int wave = tid >> 5;
    const int lane = tid & 31;
    if (wave < 4) {            // uniform per wave; EXEC all ones inside
        v8f acc = {0.f, 0.f, 0.f, 0.f, 0.f, 0.f, 0.f, 0.f};
        v2f ones; ones.x = 1.0f; ones.y = 1.0f;
        // B == all-ones  =>  D[m,n] = sum_k A[m,k] (+C), so Sum_m D[m,0] is the
        // total of all 64 A elements regardless of the A packing permutation.
        #pragma unroll
        for (int t = 0; t < 6; ++t) {   // 6 * 64 = 384 partials per residual
            v2f a;
            a.x = part[wave][t * 64 + 2 * lane];
            a.y = part[wave][t * 64 + 2 * lane + 1];
            acc = __builtin_amdgcn_wmma_f32_16x16x4_f32(
                      false, a, false, ones, (short)0, acc, false, false);
        }
        // column 0 of D: lanes 0 (M=0..7) and 16 (M=8..15), VGPRs 0..7
        float s = acc[0] + acc[1] + acc[2] + acc[3] +
                  acc[4] + acc[5] + acc[6] + acc[7];
        float tot = __shfl(s, 0, 32) + __shfl(s, 16, 32);
        if (lane == 0)
            partials[(size_t)(blockIdx.y * gridDim.x + blockIdx.x) * 4 + wave] = tot;
    }
}

// Deterministic final reduction: fixed-order double-precision sum per residual.
__global__ void pde_residual_final(const float* __restrict__ partials,
                                   float* __restrict__ out, int nblocks)
{
    const int w = threadIdx.x;
    if (w < 4) {
        double s = 0.0;
        for (int i = 0; i < nblocks; ++i)
            s += (double)partials[(size_t)i * 4 + w];
        out[w] = (float)(s / NPTS);
    }
}

extern "C" void kernel_launch(void* const* d_in, const int* in_sizes, int n_in,
                              void* d_out, int out_size, void* d_ws, size_t ws_size,
                              hipStream_t stream) {
    const float* field  = (const float*)d_in[0];   // (64,4,384,384) f32
    const float* grid   = (const float*)d_in[1];   // (64,2,384,384) f32
    const float* design = (const float*)d_in[2];   // (64,4) f32
    float* out      = (float*)d_out;               // 4 f32
    float* partials = (float*)d_ws;                // NBLOCKS*4 f32, all written each call

    dim3 g(Bdim, GRID_Y);
    pde_residual_partial<<<g, BLOCK_T, 0, stream>>>(field, grid, design, partials);
    pde_residual_final<<<1, 32, 0, stream>>>(partials, out, Bdim * GRID_Y);
}